// KMaxPooling_63617055589288
// MI455X (gfx1250) — compile-verified
//
#include <hip/hip_runtime.h>
#include <stdint.h>
#include <stddef.h>

// Problem constants (match reference: x = (64, 4096, 256) f32, k = 8, axis = 1)
#define B_      64
#define S_      4096
#define C_      256
#define KTOP    8
#define NCHUNK  8                    // sequence chunks for parallelism
#define CHUNK   (S_ / NCHUNK)        // 512
#define TILE_S  32                   // rows staged per async tile
#define NTILES  (CHUNK / TILE_S)     // 16 (even -> double-buffer parity works)
#define TILE_BYTES (TILE_S * C_ * 4) // 32768 (contiguous in global memory)

// Sorted-descending top-K register list; rare guarded insert (bubble via min/max).
__device__ __forceinline__ void topk_insert(float (&r)[KTOP], float v) {
  if (v > r[KTOP - 1]) {
    r[KTOP - 1] = v;
#pragma unroll
    for (int i = KTOP - 1; i > 0; --i) {
      float hi = fmaxf(r[i - 1], r[i]);
      float lo = fminf(r[i - 1], r[i]);
      r[i - 1] = hi;
      r[i] = lo;
    }
  }
}

// Issue one 32KB tile copy: 8 x b128 per thread, stride 4096B.
// The 24-bit inst offset is added to BOTH the LDS and global address (ISA 15.18),
// so a single (lds_addr, global_off) VGPR pair covers the whole tile.
// Data is streamed once and exceeds L2 (256MB > 192MB) -> non-temporal hint.
__device__ __forceinline__ void async_copy_tile(uint32_t lds_byte, uint32_t goff_byte,
                                                const void* gbase) {
  asm volatile(
      "global_load_async_to_lds_b128 %0, %1, %2 offset:0 th:TH_LOAD_NT\n\t"
      "global_load_async_to_lds_b128 %0, %1, %2 offset:4096 th:TH_LOAD_NT\n\t"
      "global_load_async_to_lds_b128 %0, %1, %2 offset:8192 th:TH_LOAD_NT\n\t"
      "global_load_async_to_lds_b128 %0, %1, %2 offset:12288 th:TH_LOAD_NT\n\t"
      "global_load_async_to_lds_b128 %0, %1, %2 offset:16384 th:TH_LOAD_NT\n\t"
      "global_load_async_to_lds_b128 %0, %1, %2 offset:20480 th:TH_LOAD_NT\n\t"
      "global_load_async_to_lds_b128 %0, %1, %2 offset:24576 th:TH_LOAD_NT\n\t"
      "global_load_async_to_lds_b128 %0, %1, %2 offset:28672 th:TH_LOAD_NT\n\t"
      :
      : "v"(lds_byte), "v"(goff_byte), "s"(gbase)
      : "memory");
}

// Wait for ALL outstanding async copies (used on the last tile).
__device__ __forceinline__ void wait_async0() {
  asm volatile("s_wait_asynccnt 0" ::: "memory");
}
// Async loads complete in order: counter <= 8 means the previous tile's 8 ops
// are done while the just-issued next tile's 8 remain in flight.
__device__ __forceinline__ void wait_async8() {
  asm volatile("s_wait_asynccnt 8" ::: "memory");
}

// ---------------------------------------------------------------------------
// Kernel 1: per-(batch, chunk) partial top-8 per channel, async double-buffer.
// grid = B_*NCHUNK blocks of C_ threads (thread = channel).
// part layout: [B][NCHUNK][KTOP][C]
// ---------------------------------------------------------------------------
__global__ __launch_bounds__(C_) void kmax_partial(const float* __restrict__ x,
                                                   float* __restrict__ part) {
  __shared__ float smem[2 * TILE_S * C_];  // 64 KB, double buffer

  const int c = threadIdx.x;
  const int b = blockIdx.x / NCHUNK;
  const int chunk = blockIdx.x % NCHUNK;

  // Uniform global base of this block's chunk (contiguous CHUNK*C_ floats).
  const void* gbase =
      (const void*)(x + ((size_t)b * S_ + (size_t)chunk * CHUNK) * C_);

  // This thread's copy slot: 16B at t*16, repeated at +4096B by inst offsets.
  const uint32_t voff_base = (uint32_t)threadIdx.x * 16u;
  // LDS byte offset of smem within the workgroup LDS allocation
  // (low 32 bits of the flat address of a __shared__ object == LDS offset).
  const uint32_t lds_base = (uint32_t)(uintptr_t)(&smem[0]);

  float r[KTOP];
#pragma unroll
  for (int i = 0; i < KTOP; ++i) r[i] = -__builtin_inff();

  // Prime the pipeline: tile 0 -> buffer 0.
  async_copy_tile(lds_base + voff_base, voff_base, gbase);

  for (int t = 0; t < NTILES; ++t) {
    if (t + 1 < NTILES) {
      // Issue the next tile's copy FIRST (its buffer was last read at tile
      // t-1; the barrier ending iteration t-1 makes that safe), THEN wait for
      // tile t only (<=8 outstanding): t+1's fetch overlaps t's fetch tail.
      async_copy_tile(lds_base + (uint32_t)((t + 1) & 1) * TILE_BYTES + voff_base,
                      voff_base + (uint32_t)(t + 1) * TILE_BYTES, gbase);
      wait_async8();
    } else {
      wait_async0();   // nothing left in flight behind the last tile
    }
    __syncthreads();   // all waves' copies for tile t visible in LDS

    const float* buf = &smem[(t & 1) * (TILE_S * C_)];
#pragma unroll
    for (int i = 0; i < TILE_S; ++i) {  // lane c -> bank c%64: conflict-free
      topk_insert(r, buf[i * C_ + c]);
    }
    __syncthreads();   // done reading this buffer; safe to overwrite next round
  }

  float* p = part + (((size_t)b * NCHUNK + chunk) * KTOP) * C_ + c;
#pragma unroll
  for (int j = 0; j < KTOP; ++j) p[(size_t)j * C_] = r[j];
}

// ---------------------------------------------------------------------------
// Kernel 2: merge NCHUNK partial lists per (b, c) -> sorted top-8 output.
// out layout: [B][KTOP][C]  (== reference (B, k, C))
// ---------------------------------------------------------------------------
__global__ __launch_bounds__(C_) void kmax_merge(const float* __restrict__ part,
                                                 float* __restrict__ out) {
  const int c = threadIdx.x;
  const int b = blockIdx.x;

  float r[KTOP];
#pragma unroll
  for (int i = 0; i < KTOP; ++i) r[i] = -__builtin_inff();

  const float* p = part + (size_t)b * NCHUNK * KTOP * C_ + c;
#pragma unroll
  for (int i = 0; i < NCHUNK * KTOP; ++i) topk_insert(r, p[(size_t)i * C_]);

  float* o = out + (size_t)b * KTOP * C_ + c;
#pragma unroll
  for (int j = 0; j < KTOP; ++j) o[(size_t)j * C_] = r[j];
}

// ---------------------------------------------------------------------------
// Fallback: single pass, direct coalesced global loads (used only if the
// workspace is too small for partials). Same numeric result.
// ---------------------------------------------------------------------------
__global__ __launch_bounds__(C_) void kmax_direct(const float* __restrict__ x,
                                                  float* __restrict__ out) {
  const int c = threadIdx.x;
  const int b = blockIdx.x;

  const float* p = x + (size_t)b * S_ * C_ + c;
  float r[KTOP];
#pragma unroll
  for (int i = 0; i < KTOP; ++i) r[i] = -__builtin_inff();

#pragma unroll 8
  for (int s = 0; s < S_; ++s) topk_insert(r, p[(size_t)s * C_]);

  float* o = out + (size_t)b * KTOP * C_ + c;
#pragma unroll
  for (int j = 0; j < KTOP; ++j) o[(size_t)j * C_] = r[j];
}

extern "C" void kernel_launch(void* const* d_in, const int* in_sizes, int n_in,
                              void* d_out, int out_size, void* d_ws, size_t ws_size,
                              hipStream_t stream) {
  (void)in_sizes; (void)n_in; (void)out_size;
  const float* x = (const float*)d_in[0];
  float* out = (float*)d_out;

  const size_t need = (size_t)B_ * NCHUNK * KTOP * C_ * sizeof(float);  // 4 MB
  if (d_ws != nullptr && ws_size >= need) {
    float* part = (float*)d_ws;
    kmax_partial<<<dim3(B_ * NCHUNK), dim3(C_), 0, stream>>>(x, part);
    kmax_merge<<<dim3(B_), dim3(C_), 0, stream>>>(part, out);
  } else {
    kmax_direct<<<dim3(B_), dim3(C_), 0, stream>>>(x, out);
  }
}